// Char_30322469110372
// MI455X (gfx1250) — compile-verified
//
#include <hip/hip_runtime.h>
#include <hip/hip_bf16.h>

typedef float v2f __attribute__((ext_vector_type(2)));
typedef float v8f __attribute__((ext_vector_type(8)));

#define CDIM  100
#define LSLOT 7
#define RSLOT 10
#define WDIM  300
#define NEGV  -1e30f

__device__ __forceinline__ float wave_sum(float v) {
  #pragma unroll
  for (int m = 16; m; m >>= 1) v += __shfl_xor(v, m, 32);
  return v;
}

// score = (ce_raw[idx] / ||ce_raw[idx]||) . vrow   (vrow already includes q-normalization)
__device__ __forceinline__ float score_slot(int idx, const float* vrow,
                                            const float* __restrict__ ce_raw, int lane) {
  if (idx < 0) return NEGV;                       // '<pad>' slot -> masked
  const float* row = ce_raw + (long)idx * CDIM;
  float dot = 0.f, ss = 0.f;
  for (int c = lane; c < CDIM; c += 32) {
    float x = row[c];
    dot = fmaf(x, vrow[c], dot);
    ss  = fmaf(x, x, ss);
  }
  #pragma unroll
  for (int m = 16; m; m >>= 1) {                  // butterfly: all lanes get totals
    dot += __shfl_xor(dot, m, 32);
    ss  += __shfl_xor(ss,  m, 32);
  }
  return dot / fmaxf(sqrtf(ss), 1e-12f);
}

__global__ __launch_bounds__(64) void fused_gated_attn_kernel(
    const float* __restrict__ ce_raw, const float* __restrict__ wvec,
    const float* __restrict__ lW,     const float* __restrict__ rW,
    const float* __restrict__ gL_w,   const float* __restrict__ gL_b,
    const int* __restrict__ qidx,
    const int* __restrict__ lixs_c,   const int* __restrict__ rixs_c,
    const int* __restrict__ lixs_w,   const int* __restrict__ rixs_w,
    float* __restrict__ out)
{
  __shared__ float q_lds[32][104];        // normalized Q rows (padded)
  __shared__ float v_lds[32][2][112];     // V = Qn @ W^T, per side, N padded to 112

  const int lane  = threadIdx.x & 31;
  const int wv    = threadIdx.x >> 5;     // wave 0..1; each wave owns 16 batch rows
  const int bbase = blockIdx.x * 32 + wv * 16;
  const int hi = lane >> 4, ln = lane & 15;

  // ---- phase 0: gather + L2-normalize 16 query rows into LDS (coalesced per row)
  for (int r = 0; r < 16; ++r) {
    long row = qidx[bbase + r];
    float vals[4]; float ss = 0.f;
    #pragma unroll
    for (int i = 0; i < 4; ++i) {
      int c = lane + 32 * i;
      vals[i] = (c < CDIM) ? ce_raw[row * CDIM + c] : 0.f;
      ss = fmaf(vals[i], vals[i], ss);
    }
    ss = wave_sum(ss);
    float inv = 1.f / fmaxf(sqrtf(ss), 1e-12f);
    #pragma unroll
    for (int i = 0; i < 4; ++i) {
      int c = lane + 32 * i;
      if (c < CDIM) q_lds[wv * 16 + r][c] = vals[i] * inv;
    }
  }
  __syncthreads();

  // ---- phase 1: V[b,c] = sum_d Qn[b,d] * W[c,d] via V_WMMA_F32_16X16X4_F32
  // A(16x4):  lane(hi,ln), a[v] = Qn[M=ln, K=d0+v], d0 = 4k + 2*hi
  // B(4x16):  b[v] = W^T[K=d0+v, N=ln+16*nt] = W[cout, d0+v]
  // D(16x16): acc[r] = V[M = r + 8*hi, N = ln]
  for (int side = 0; side < 2; ++side) {
    const float* W = side ? rW : lW;
    for (int nt = 0; nt < 7; ++nt) {
      int cout = nt * 16 + ln;
      v8f acc = {};
      for (int k = 0; k < 25; ++k) {
        int d0 = k * 4 + 2 * hi;
        v2f a, bfrag;
        a[0] = q_lds[wv * 16 + ln][d0];
        a[1] = q_lds[wv * 16 + ln][d0 + 1];
        if (cout < CDIM) {
          bfrag[0] = W[cout * CDIM + d0];
          bfrag[1] = W[cout * CDIM + d0 + 1];
        } else {
          bfrag[0] = 0.f; bfrag[1] = 0.f;        // zero-pad N=100..111
        }
        acc = __builtin_amdgcn_wmma_f32_16x16x4_f32(
            /*neg_a=*/false, a, /*neg_b=*/false, bfrag,
            /*c_mod=*/(short)0, acc, /*reuse_a=*/false, /*reuse_b=*/false);
      }
      #pragma unroll
      for (int r = 0; r < 8; ++r)
        v_lds[wv * 16 + r + 8 * hi][side][nt * 16 + ln] = acc[r];
    }
  }
  __syncthreads();

  // ---- phase 2: per batch row: scores -> softmax -> gate -> gathered emb
  for (int bb = 0; bb < 16; ++bb) {
    const int b  = bbase + bb;
    const int bl = wv * 16 + bb;

    float la[LSLOT], ra[RSLOT];
    #pragma unroll
    for (int t = 0; t < LSLOT; ++t)
      la[t] = score_slot(lixs_c[b * LSLOT + t], &v_lds[bl][0][0], ce_raw, lane);
    #pragma unroll
    for (int t = 0; t < RSLOT; ++t)
      ra[t] = score_slot(rixs_c[b * RSLOT + t], &v_lds[bl][1][0], ce_raw, lane);

    // softmax over 7 (left): masked slots are NEGV -> expf underflows to exactly 0
    float mx = la[0];
    #pragma unroll
    for (int t = 1; t < LSLOT; ++t) mx = fmaxf(mx, la[t]);
    float s = 0.f;
    #pragma unroll
    for (int t = 0; t < LSLOT; ++t) { la[t] = expf(la[t] - mx); s += la[t]; }
    float rs = 1.f / s;
    #pragma unroll
    for (int t = 0; t < LSLOT; ++t) la[t] *= rs;

    // softmax over 10 (right)
    mx = ra[0];
    #pragma unroll
    for (int t = 1; t < RSLOT; ++t) mx = fmaxf(mx, ra[t]);
    s = 0.f;
    #pragma unroll
    for (int t = 0; t < RSLOT; ++t) { ra[t] = expf(ra[t] - mx); s += ra[t]; }
    rs = 1.f / s;
    #pragma unroll
    for (int t = 0; t < RSLOT; ++t) ra[t] *= rs;

    // gate = softmax( concat(la,ra) @ gL_w.T + gL_b ) over 2
    float z0 = gL_b[0], z1 = gL_b[1];
    #pragma unroll
    for (int t = 0; t < LSLOT; ++t) {
      z0 = fmaf(la[t], gL_w[t],                z0);
      z1 = fmaf(la[t], gL_w[(LSLOT + RSLOT) + t], z1);
    }
    #pragma unroll
    for (int t = 0; t < RSLOT; ++t) {
      z0 = fmaf(ra[t], gL_w[LSLOT + t],                       z0);
      z1 = fmaf(ra[t], gL_w[(LSLOT + RSLOT) + LSLOT + t],     z1);
    }
    float gm = fmaxf(z0, z1);
    float e0 = expf(z0 - gm), e1 = expf(z1 - gm);
    float gi = 1.f / (e0 + e1);
    float g0 = e0 * gi, g1 = e1 * gi;

    // emb: e[b,j] = g0 * sum_t la[t]*wvec[lixs_w] + g1 * sum_t ra[t]*wvec[rixs_w]
    for (int j = lane; j < WDIM; j += 32) {
      float accL = 0.f, accR = 0.f;
      #pragma unroll
      for (int t = 0; t < LSLOT; ++t) {
        int iw = lixs_w[b * LSLOT + t];
        if (iw >= 0) accL = fmaf(la[t], wvec[(long)iw * WDIM + j], accL);
      }
      #pragma unroll
      for (int t = 0; t < RSLOT; ++t) {
        int iw = rixs_w[b * RSLOT + t];
        if (iw >= 0) accR = fmaf(ra[t], wvec[(long)iw * WDIM + j], accR);
      }
      out[(long)b * WDIM + j] = g0 * accL + g1 * accR;
    }
  }
}

extern "C" void kernel_launch(void* const* d_in, const int* in_sizes, int n_in,
                              void* d_out, int out_size, void* d_ws, size_t ws_size,
                              hipStream_t stream) {
  const float* ce_raw = (const float*)d_in[0];
  const float* wvec   = (const float*)d_in[1];
  const float* lW     = (const float*)d_in[2];
  const float* rW     = (const float*)d_in[3];
  const float* gL_w   = (const float*)d_in[4];
  const float* gL_b   = (const float*)d_in[5];
  const int*   qidx   = (const int*)d_in[6];
  const int*   lixs_c = (const int*)d_in[7];
  const int*   rixs_c = (const int*)d_in[8];
  const int*   lixs_w = (const int*)d_in[9];
  const int*   rixs_w = (const int*)d_in[10];

  const int B = in_sizes[6];          // 16384
  const int blocks = B / 32;          // 2 waves/block, 16 batch rows per wave
  fused_gated_attn_kernel<<<blocks, 64, 0, stream>>>(
      ce_raw, wvec, lW, rW, gL_w, gL_b,
      qidx, lixs_c, rixs_c, lixs_w, rixs_w, (float*)d_out);
}